// SX_6828998000762
// MI455X (gfx1250) — compile-verified
//
#include <hip/hip_runtime.h>

// 16 qudits of dim 3 (3^16 f32 amplitudes, re+im). Gate = real 9x9 matrix M on
// axes (2,5): identity except M[3][4]=M[4][3]=1, row/col 5 zero, diag 3,4,5 zero.
// 0 FLOPs/byte -> pure HBM streaming (~650 MB, ~28 us @ 23.3 TB/s).  We run the
// contraction on the matrix pipe anyway: V_WMMA_F32_16X16X4_F32 is fp32 and
// BIT-EXACT for a 0/1 matrix (each dot product has <=1 nonzero term), so the
// matrix units come for free while loads/stores stay coalesced + non-temporal.

typedef float v2f __attribute__((ext_vector_type(2)));
typedef float v8f __attribute__((ext_vector_type(8)));

constexpr unsigned kN    = 43046721u; // 3^16 state elements
constexpr unsigned kS2   = 1594323u;  // 3^13 stride of qudit axis 2 (digit a)
constexpr unsigned kS5   = 59049u;    // 3^10 stride of qudit axis 5 (digit b)
constexpr unsigned kSO01 = 4782969u;  // 3^14 stride of combined axes 0-1
constexpr unsigned kSO34 = 177147u;   // 3^11 stride of combined axes 3-4
constexpr unsigned kCols = 4782969u;  // 3^14 non-gate "columns"
constexpr unsigned kTiles = (kCols + 15u) / 16u; // 298936 wave tiles

// byte-free offset of gate index k = 3a+b relative to a column base
constexpr unsigned gOff(unsigned k) { return (k / 3u) * kS2 + (k % 3u) * kS5; }

// Gate matrix entry A[m][k] (0/1), padded with zeros for m,k >= 9.
__device__ __forceinline__ float aval(unsigned m, unsigned k) {
  bool one = (m == k) ? (m <= 2u || (m >= 6u && m <= 8u))
                      : ((m == 3u && k == 4u) || (m == 4u && k == 3u));
  return one ? 1.0f : 0.0f;
}

// One B-operand VGPR: low lanes carry K=KLO, high lanes K=KHI.
// Invalid K (padding, or the dead K=5 gate column) -> load a cached dup of
// offset 0 and select 0.0f (never feeds uninitialized/NaN data into the MAC).
template <unsigned KLO, unsigned KHI>
__device__ __forceinline__ float bload(const float* __restrict__ p,
                                       unsigned c, bool hi) {
  constexpr bool vlo = (KLO <= 8u) && (KLO != 5u);
  constexpr bool vhi = (KHI <= 8u) && (KHI != 5u);
  if (!vlo && !vhi) return 0.0f;                    // no load at all
  constexpr unsigned olo = vlo ? gOff(KLO) : 0u;
  constexpr unsigned ohi = vhi ? gOff(KHI) : 0u;
  float v = p[c + (hi ? ohi : olo)];
  return (hi ? vhi : vlo) ? v : 0.0f;
}

__global__ __launch_bounds__(256) void sx_wmma_kernel(
    const float* __restrict__ qr, const float* __restrict__ qi,
    float* __restrict__ outr, float* __restrict__ outi)
{
  unsigned lane = threadIdx.x & 31u;
  unsigned wave = threadIdx.x >> 5u;
  unsigned tile = blockIdx.x * 8u + wave;
  if (tile >= kTiles) return;            // wave-uniform: EXEC all-1s below

  bool     hi = lane >= 16u;
  unsigned m  = lane & 15u;              // A row / B/D column / column id
  unsigned n  = tile * 16u + m;          // global column for this lane
  bool   vcol = n < kCols;
  unsigned nv = vcol ? n : (kCols - 1u); // clamp partial tile for loads

  // column base: flat index with a = b = 0
  unsigned low  = nv % kS5;
  unsigned rest = nv / kS5;
  unsigned o34  = rest % 9u;
  unsigned o01  = rest / 9u;
  unsigned c    = o01 * kSO01 + o34 * kSO34 + low;

  // A tiles (constant 0/1 gate matrix, synthesized in registers)
  unsigned kb = hi ? 2u : 0u;
  v2f a0 = { aval(m, 0u + kb), aval(m, 1u + kb) };   // K 0..3
  v2f a1 = { aval(m, 4u + kb), aval(m, 5u + kb) };   // K 4..7
  v2f a2 = { aval(m, 8u + kb), aval(m, 9u + kb) };   // K 8..11 (pad)

  // B tiles: coalesced b32 gathers of the 8 live gate-input rows
  v2f br0 = { bload<0u, 2u >(qr, c, hi), bload<1u, 3u >(qr, c, hi) };
  v2f br1 = { bload<4u, 6u >(qr, c, hi), bload<5u, 7u >(qr, c, hi) };
  v2f br2 = { bload<8u, 10u>(qr, c, hi), bload<9u, 11u>(qr, c, hi) };
  v2f bi0 = { bload<0u, 2u >(qi, c, hi), bload<1u, 3u >(qi, c, hi) };
  v2f bi1 = { bload<4u, 6u >(qi, c, hi), bload<5u, 7u >(qi, c, hi) };
  v2f bi2 = { bload<8u, 10u>(qi, c, hi), bload<9u, 11u>(qi, c, hi) };

  // D = A x B, K chained through the accumulator (exact: 0/1 matrix, fp32 MAC)
  v8f dr = {};
  dr = __builtin_amdgcn_wmma_f32_16x16x4_f32(false, a0, false, br0, (short)0, dr, false, false);
  dr = __builtin_amdgcn_wmma_f32_16x16x4_f32(false, a1, false, br1, (short)0, dr, false, false);
  dr = __builtin_amdgcn_wmma_f32_16x16x4_f32(false, a2, false, br2, (short)0, dr, false, false);
  v8f di = {};
  di = __builtin_amdgcn_wmma_f32_16x16x4_f32(false, a0, false, bi0, (short)0, di, false, false);
  di = __builtin_amdgcn_wmma_f32_16x16x4_f32(false, a1, false, bi1, (short)0, di, false, false);
  di = __builtin_amdgcn_wmma_f32_16x16x4_f32(false, a2, false, bi2, (short)0, di, false, false);

  // Scatter rows 0..8 of D (VGPR g holds rows g and g+8); rows 9..15 are pad.
  if (vcol) {
    unsigned o0 = hi ? gOff(8u) : gOff(0u);          // VGPR0: rows 0 and 8
    __builtin_nontemporal_store(dr[0], outr + c + o0);
    __builtin_nontemporal_store(di[0], outi + c + o0);
    if (!hi) {                                       // rows 1..7 (incl. zero row 5)
#pragma unroll
      for (unsigned g = 1u; g < 8u; ++g) {
        __builtin_nontemporal_store(dr[g], outr + c + gOff(g));
        __builtin_nontemporal_store(di[g], outi + c + gOff(g));
      }
    }
  }
}

extern "C" void kernel_launch(void* const* d_in, const int* in_sizes, int n_in,
                              void* d_out, int out_size, void* d_ws, size_t ws_size,
                              hipStream_t stream) {
  const float* qr = (const float*)d_in[0]; // qs_real
  const float* qi = (const float*)d_in[1]; // qs_imag
  float* outr = (float*)d_out;             // out_re
  float* outi = outr + kN;                 // out_im (tuple outputs concatenated)

  unsigned blocks = (kTiles + 7u) / 8u;    // 8 waves (tiles) per 256-thread block
  sx_wmma_kernel<<<blocks, 256, 0, stream>>>(qr, qi, outr, outi);
}